// RUL_loss_mixed_20177756356930
// MI455X (gfx1250) — compile-verified
//
#include <hip/hip_runtime.h>
#include <math.h>

typedef __attribute__((ext_vector_type(2))) float v2f;
typedef __attribute__((ext_vector_type(8))) float v8f;

#define THREADS 256
#define BLOCKS  1024

// Reduce two 256-float LDS arrays simultaneously with V_WMMA_F32_16X16X4_F32.
// B = all-ones (layout independent) => D[m][n] = rowsum(A, m) + C[m][n].
// A rows 0-7 carry scoring partials, rows 8-15 carry squared-diff partials.
// Documented fp32 A 16x4 layout: lanes 0-15 hold K=0 (a.x) / K=1 (a.y) for M=lane;
// lanes 16-31 hold K=2 / K=3 for M=lane-16.
// Result (every lane of wave 0): lanes 0-15 -> total(lds_sc), lanes 16-31 -> total(lds_sq).
__device__ __forceinline__ float wmma_block_reduce2(const float* lds_sc, const float* lds_sq) {
    const int lane = threadIdx.x;            // caller guarantees wave 0 only (0..31)
    const int m    = lane & 15;              // matrix row this lane feeds
    const int kb   = (lane >> 4) << 1;       // 0 for lanes 0-15, 2 for lanes 16-31
    const int row  = m & 7;                  // row within the 8-row half
    const float* src = (m < 8) ? lds_sc : lds_sq;

    v8f c = {0.f, 0.f, 0.f, 0.f, 0.f, 0.f, 0.f, 0.f};
    v2f ones; ones.x = 1.0f; ones.y = 1.0f;

#pragma unroll
    for (int chunk = 0; chunk < THREADS / 32; ++chunk) {
        const int base = chunk * 32 + row * 4 + kb;
        v2f a; a.x = src[base]; a.y = src[base + 1];
        // D = A x ones + C : accumulates 32 scoring + 32 sq values per issue
        c = __builtin_amdgcn_wmma_f32_16x16x4_f32(
                /*neg_a=*/false, a, /*neg_b=*/false, ones,
                /*c_mod=*/(short)0, c, /*reuse_a=*/false, /*reuse_b=*/false);
    }
    return c[0] + c[1] + c[2] + c[3] + c[4] + c[5] + c[6] + c[7];
}

__device__ __forceinline__ void accum_elem(float p, float l, float& sc, float& sq) {
    const float d = p - l;
    const float s = (d < 0.f) ? (-1.0f / 13.0f) : (1.0f / 10.0f);
    sc += expf(d * s) - 1.0f;     // matches jnp.where(d<0, exp(-d/13), exp(d/10)) - 1
    sq = fmaf(d, d, sq);
}

__global__ void rul_partial_kernel(const float* __restrict__ pred,
                                   const float* __restrict__ lab,
                                   float* __restrict__ ws, int n) {
    __shared__ float lds_sc[THREADS];
    __shared__ float lds_sq[THREADS];

    const int tid    = blockIdx.x * blockDim.x + threadIdx.x;
    const int stride = gridDim.x * blockDim.x;
    const int nvec   = n >> 2;

    const float4* p4 = (const float4*)pred;
    const float4* l4 = (const float4*)lab;

    float sc = 0.f, sq = 0.f;
    for (int i = tid; i < nvec; i += stride) {       // b128 streaming loads
        const float4 p = p4[i];
        const float4 l = l4[i];
        accum_elem(p.x, l.x, sc, sq);
        accum_elem(p.y, l.y, sc, sq);
        accum_elem(p.z, l.z, sc, sq);
        accum_elem(p.w, l.w, sc, sq);
    }
    if (tid == 0) {                                  // scalar tail (n % 4)
        for (int i = (nvec << 2); i < n; ++i) accum_elem(pred[i], lab[i], sc, sq);
    }

    lds_sc[threadIdx.x] = sc;
    lds_sq[threadIdx.x] = sq;
    __syncthreads();

    if (threadIdx.x < 32) {                          // whole wave 0: EXEC all-ones for WMMA
        const float r = wmma_block_reduce2(lds_sc, lds_sq);
        if (threadIdx.x == 0)  ws[2 * blockIdx.x + 0] = r;   // block scoring sum
        if (threadIdx.x == 16) ws[2 * blockIdx.x + 1] = r;   // block sq sum
    }
}

__global__ void rul_final_kernel(const float* __restrict__ ws,
                                 float* __restrict__ out,
                                 int nblocks, float inv_n) {
    __shared__ float lds_sc[THREADS];
    __shared__ float lds_sq[THREADS];

    float sc = 0.f, sq = 0.f;
    for (int i = threadIdx.x; i < nblocks; i += THREADS) {
        sc += ws[2 * i + 0];
        sq += ws[2 * i + 1];
    }
    lds_sc[threadIdx.x] = sc;
    lds_sq[threadIdx.x] = sq;
    __syncthreads();

    if (threadIdx.x < 32) {
        const float r      = wmma_block_reduce2(lds_sc, lds_sq);
        const float sc_tot = __shfl(r, 0, 32);
        const float sq_tot = __shfl(r, 16, 32);
        if (threadIdx.x == 0)
            out[0] = 0.5f * sc_tot + 0.5f * sqrtf(sq_tot * inv_n);
    }
}

extern "C" void kernel_launch(void* const* d_in, const int* in_sizes, int n_in,
                              void* d_out, int out_size, void* d_ws, size_t ws_size,
                              hipStream_t stream) {
    const float* pred = (const float*)d_in[0];
    const float* lab  = (const float*)d_in[1];
    float* out = (float*)d_out;
    float* ws  = (float*)d_ws;

    const int n    = in_sizes[0];
    const int nvec = n >> 2;
    int blocks = (nvec + THREADS - 1) / THREADS;
    if (blocks > BLOCKS) blocks = BLOCKS;
    if (blocks < 1) blocks = 1;

    rul_partial_kernel<<<blocks, THREADS, 0, stream>>>(pred, lab, ws, n);
    rul_final_kernel<<<1, THREADS, 0, stream>>>(ws, out, blocks, 1.0f / (float)n);
}